// STBlock_16243566313773
// MI455X (gfx1250) — compile-verified
//
#include <hip/hip_runtime.h>
#include <math.h>

#define B_   4
#define T_   12
#define N_   325
#define D_   32
#define H_   8
#define ROWS_ (B_*T_*N_)       /* 15600 tokens */
#define SCW  336               /* 21 t-tiles * 16, padded score row width */
#define SCALE_ 0.5f            /* 1/sqrt(4) */

typedef float v2f __attribute__((ext_vector_type(2)));
typedef float v8f __attribute__((ext_vector_type(8)));

__device__ __forceinline__ v8f wmma4(v2f a, v2f b, v8f c) {
  // D(16x16,f32) = A(16x4,f32) * B(4x16,f32) + C
  return __builtin_amdgcn_wmma_f32_16x16x4_f32(false, a, false, b, (short)0, c, false, false);
}

// ---------------- embedding: e = relu(W2 relu(W1 x + b1) + b2) ----------------
__global__ void k_embed(const float* __restrict__ x,
                        const float* __restrict__ w1, const float* __restrict__ b1,
                        const float* __restrict__ w2, const float* __restrict__ b2,
                        float* __restrict__ e) {
  int tok = blockIdx.x * blockDim.x + threadIdx.x;
  if (tok >= ROWS_) return;
  float xv = x[tok];
  float h[16];
#pragma unroll
  for (int j = 0; j < 16; ++j) h[j] = fmaxf(0.f, w1[j] * xv + b1[j]);
#pragma unroll
  for (int c = 0; c < D_; ++c) {
    float s = b2[c];
#pragma unroll
    for (int j = 0; j < 16; ++j) s += w2[c * 16 + j] * h[j];
    e[(size_t)tok * D_ + c] = fmaxf(0.f, s);
  }
}

// ---------------- layout transposes (B,T,N,32) <-> (B,N,T,32) ----------------
__global__ void k_btn_to_bnt(const float* __restrict__ src, float* __restrict__ dst) {
  int i = blockIdx.x * blockDim.x + threadIdx.x;
  if (i >= ROWS_ * D_) return;
  int c = i % D_; int r = i / D_;
  int n = r % N_; int bt = r / N_;
  int t = bt % T_; int b = bt / T_;
  dst[((size_t)((b * N_ + n) * T_ + t)) * D_ + c] = src[i];
}
__global__ void k_bnt_to_btn(const float* __restrict__ src, float* __restrict__ dst) {
  int i = blockIdx.x * blockDim.x + threadIdx.x;
  if (i >= ROWS_ * D_) return;
  int c = i % D_; int r = i / D_;
  int t = r % T_; int bn = r / T_;
  int n = bn % N_; int b = bn / N_;
  dst[((size_t)((b * T_ + t) * N_ + n)) * D_ + c] = src[i];
}

// ---------------- generic WMMA linear: Y = act(X @ W^T + b) ----------------
// X rows have 32 channels from in0; if I==64, channels 32..63 come from in1.
// ACT: 0=none 1=relu 2=sigmoid. GATHER: out row r=(b*N+n) reads (b*T+T-1)*N+n.
// Out-of-range tile rows are clamped (their D rows are masked at store).
template<int I, int ACT, int GATHER>
__global__ void k_lin(const float* __restrict__ in0, const float* __restrict__ in1,
                      const float* __restrict__ W,  const float* __restrict__ bias,
                      float* __restrict__ out, int rows) {
  __shared__ float Wl[32 * I];
  int l = threadIdx.x; int hl = l >> 4; int nn = l & 15; int kbo = hl * 2;
  __builtin_prefetch(W, 0, 1);
  // cooperative stage of W (O=32 x I, row-major) into LDS, 16B per lane per iter
#pragma unroll
  for (int i = l * 4; i < 32 * I; i += 128) {
    const float4 w4 = *(const float4*)(W + i);
    Wl[i] = w4.x; Wl[i + 1] = w4.y; Wl[i + 2] = w4.z; Wl[i + 3] = w4.w;
  }
  __syncthreads();
  int s0 = blockIdx.x * 16;
  int row = s0 + nn;
  int rowc = (row < rows) ? row : (rows - 1);
  int inrow = rowc;
  if (GATHER) { int b = rowc / N_; int n = rowc % N_; inrow = (b * T_ + (T_ - 1)) * N_ + n; }
  const float* arow0 = in0 + (size_t)inrow * D_;
  const float* arow1 = (I == 64) ? (in1 + (size_t)inrow * D_) : arow0;
  constexpr int NKS = I / 4;
  v2f av[NKS];
#pragma unroll
  for (int ks = 0; ks < NKS; ++ks) {
    int kg = ks * 4 + kbo;
    const float* p = (kg < D_) ? (arow0 + kg) : (arow1 + (kg - D_));
    av[ks].x = p[0]; av[ks].y = p[1];
  }
#pragma unroll
  for (int ct = 0; ct < 2; ++ct) {
    v8f c = {};
#pragma unroll
    for (int ks = 0; ks < NKS; ++ks) {
      int kg = ks * 4 + kbo;
      v2f b; b.x = Wl[(ct * 16 + nn) * I + kg]; b.y = Wl[(ct * 16 + nn) * I + kg + 1];
      c = wmma4(av[ks], b, c);
    }
    float bb = bias[ct * 16 + nn];
#pragma unroll
    for (int r = 0; r < 8; ++r) {
      int orow = s0 + r + 8 * hl;
      if (orow < rows) {
        float v = c[r] + bb;
        if (ACT == 1) v = fmaxf(v, 0.f);
        else if (ACT == 2) v = 1.f / (1.f + expf(-v));
        out[(size_t)orow * D_ + ct * 16 + nn] = v;
      }
    }
  }
}

// ---------------- attention core: softmax((Q K^T * s) [*adj | mask]) V ----------------
// grid: (s_tiles, H, BB); one wave per (bb, s-tile, head).
// ADJ: multiply scores by adj[s,t]; CAUSAL: -inf where t > s.
// All loads clamped/unconditional; masking is pure select arithmetic (no exec
// divergence around loads). V head-slice staged in LDS via aligned float4.
template<int ADJ, int CAUSAL>
__global__ void k_attn(const float* __restrict__ q, const float* __restrict__ k,
                       const float* __restrict__ v, float* __restrict__ o,
                       const float* __restrict__ adj, int S) {
  __shared__ float sc[16 * SCW];
  __shared__ float Vl[SCW * 4];
  int st = blockIdx.x, h = blockIdx.y, bb = blockIdx.z;
  int ntt = (S + 15) >> 4;
  int l = threadIdx.x; int hl = l >> 4; int nn = l & 15; int kbo = hl * 2;
  int s0 = st * 16;
  // stage V[bb, 0..S, h*4..h*4+3] into LDS (16B-aligned vector loads)
  for (int t = l; t < S; t += 32) {
    const float4 v4 = *(const float4*)(v + ((size_t)bb * S + t) * D_ + h * 4);
    Vl[t * 4 + 0] = v4.x; Vl[t * 4 + 1] = v4.y; Vl[t * 4 + 2] = v4.z; Vl[t * 4 + 3] = v4.w;
  }
  v2f a;
  {
    int srowc = s0 + nn; if (srowc > S - 1) srowc = S - 1;
    const float* p = q + ((size_t)bb * S + srowc) * D_ + h * 4 + kbo;
    a.x = p[0]; a.y = p[1];
  }
  for (int tt = 0; tt < ntt; ++tt) {
    int t0 = tt * 16;
    int trowc = t0 + nn; if (trowc > S - 1) trowc = S - 1;
    const float* p = k + ((size_t)bb * S + trowc) * D_ + h * 4 + kbo;
    v2f b; b.x = p[0]; b.y = p[1];
    v8f c = {};
    c = wmma4(a, b, c);
    int t  = t0 + nn;
    int tc = (t < S) ? t : (S - 1);
#pragma unroll
    for (int r = 0; r < 8; ++r) {
      int m = r + 8 * hl; int srow = s0 + m;
      int sclamp = (srow < S) ? srow : (S - 1);
      float adjv = 1.f;
      if (ADJ) adjv = adj[(size_t)sclamp * S + tc];           // unconditional load
      bool ok = (srow < S) & (t < S);
      if (CAUSAL) ok = ok & (t <= srow);
      float val = ok ? (c[r] * SCALE_ * adjv) : -1e30f;        // pure select
      sc[m * SCW + t] = val;
    }
  }
  __syncthreads();
  if (l < 16) {                       // softmax per row
    float* rowp = sc + l * SCW;
    float mx = -1e30f;
    for (int t = 0; t < S; ++t) mx = fmaxf(mx, rowp[t]);
    float sum = 0.f;
    for (int t = 0; t < S; ++t) { float e = expf(rowp[t] - mx); rowp[t] = e; sum += e; }
    float inv = 1.f / sum;
    for (int t = 0; t < S; ++t) rowp[t] *= inv;
  }
  __syncthreads();
  // A @ V from LDS: lane handles (row nn, head-dims hl*2, hl*2+1)
  int srow = s0 + nn;
  if (srow < S) {
    float acc0 = 0.f, acc1 = 0.f;
    const float* rowp = sc + nn * SCW;
    for (int t = 0; t < S; ++t) {
      float p = rowp[t];
      acc0 += p * Vl[t * 4 + kbo]; acc1 += p * Vl[t * 4 + kbo + 1];
    }
    float* op = o + ((size_t)bb * S + srow) * D_ + h * 4 + kbo;
    op[0] = acc0; op[1] = acc1;
  }
}

// ---------------- out-proj + residual + LayerNorm ----------------
__global__ void k_projln(const float* __restrict__ xin, const float* __restrict__ oc,
                         const float* __restrict__ Wo, const float* __restrict__ bo,
                         const float* __restrict__ g,  const float* __restrict__ be,
                         float* __restrict__ out, int rows) {
  __shared__ float Wl[32 * D_];
  __shared__ float P[16][D_];
  int l = threadIdx.x; int hl = l >> 4; int nn = l & 15; int kbo = hl * 2;
#pragma unroll
  for (int i = l * 4; i < 32 * D_; i += 128) {
    const float4 w4 = *(const float4*)(Wo + i);
    Wl[i] = w4.x; Wl[i + 1] = w4.y; Wl[i + 2] = w4.z; Wl[i + 3] = w4.w;
  }
  __syncthreads();
  int s0 = blockIdx.x * 16;
  int row = s0 + nn;
  int rowc = (row < rows) ? row : (rows - 1);
  const float* arow = oc + (size_t)rowc * D_;
  v2f av[8];
#pragma unroll
  for (int ks = 0; ks < 8; ++ks) { int kg = ks * 4 + kbo; av[ks].x = arow[kg]; av[ks].y = arow[kg + 1]; }
#pragma unroll
  for (int ct = 0; ct < 2; ++ct) {
    v8f c = {};
#pragma unroll
    for (int ks = 0; ks < 8; ++ks) {
      int kg = ks * 4 + kbo;
      v2f b; b.x = Wl[(ct * 16 + nn) * D_ + kg]; b.y = Wl[(ct * 16 + nn) * D_ + kg + 1];
      c = wmma4(av[ks], b, c);
    }
    float bb = bo[ct * 16 + nn];
#pragma unroll
    for (int r = 0; r < 8; ++r) P[r + 8 * hl][ct * 16 + nn] = c[r] + bb;
  }
  __syncthreads();
  if (l < 16) {
    int r2 = s0 + l;
    if (r2 < rows) {
      float s[D_]; float mean = 0.f;
#pragma unroll
      for (int c = 0; c < D_; ++c) { s[c] = xin[(size_t)r2 * D_ + c] + P[l][c]; mean += s[c]; }
      mean *= (1.f / (float)D_);
      float var = 0.f;
#pragma unroll
      for (int c = 0; c < D_; ++c) { float d = s[c] - mean; var += d * d; }
      var *= (1.f / (float)D_);
      float inv = rsqrtf(var + 1e-5f);
#pragma unroll
      for (int c = 0; c < D_; ++c)
        out[(size_t)r2 * D_ + c] = (s[c] - mean) * inv * g[c] + be[c];
    }
  }
}

// ---------------- gating: hst = z*hdt + (1-z)*hds ----------------
__global__ void k_gate(const float* __restrict__ z, const float* __restrict__ hdt,
                       const float* __restrict__ hds, float* __restrict__ hst) {
  int i = blockIdx.x * blockDim.x + threadIdx.x;
  if (i >= ROWS_ * D_) return;
  float zz = z[i];
  hst[i] = zz * hdt[i] + (1.f - zz) * hds[i];
}

// ---------------- causal dilated conv (k=3) via 3 accumulated shifted GEMMs ----
// channel-last (B,T,N,32); w layout (O=32, I=32, 3, 1); out = relu(conv + bias)
// Left padding handled branchlessly: clamp address, multiply by 0/1 mask.
__global__ void k_conv(const float* __restrict__ in, const float* __restrict__ w,
                       const float* __restrict__ bias, float* __restrict__ out, int dil) {
  __shared__ float Wl[32 * 32 * 3];
  int l = threadIdx.x; int hl = l >> 4; int nn = l & 15; int kbo = hl * 2;
#pragma unroll
  for (int i = l * 4; i < 32 * 32 * 3; i += 128) {
    const float4 w4 = *(const float4*)(w + i);
    Wl[i] = w4.x; Wl[i + 1] = w4.y; Wl[i + 2] = w4.z; Wl[i + 3] = w4.w;
  }
  __syncthreads();
  int s0 = blockIdx.x * 16;
  int row = s0 + nn;
  int rowc = (row < ROWS_) ? row : (ROWS_ - 1);
  int b = rowc / (T_ * N_); int rem = rowc % (T_ * N_); int t = rem / N_; int n = rem % N_;
  v8f c0 = {}; v8f c1 = {};
#pragma unroll
  for (int kt = 0; kt < 3; ++kt) {
    int tp = t - (2 - kt) * dil;
    int tpc = (tp > 0) ? tp : 0;
    const float* arow = in + ((size_t)((b * T_ + tpc) * N_ + n)) * D_;
    float msk = (tp >= 0) ? 1.f : 0.f;
#pragma unroll
    for (int ks = 0; ks < 8; ++ks) {
      int kg = ks * 4 + kbo;
      v2f a; a.x = msk * arow[kg]; a.y = msk * arow[kg + 1];
      v2f b0; b0.x = Wl[(nn * D_ + kg) * 3 + kt];
      b0.y = Wl[(nn * D_ + kg + 1) * 3 + kt];
      c0 = wmma4(a, b0, c0);
      v2f b1; b1.x = Wl[((16 + nn) * D_ + kg) * 3 + kt];
      b1.y = Wl[((16 + nn) * D_ + kg + 1) * 3 + kt];
      c1 = wmma4(a, b1, c1);
    }
  }
  float bb0 = bias[nn], bb1 = bias[16 + nn];
#pragma unroll
  for (int r = 0; r < 8; ++r) {
    int orow = s0 + r + 8 * hl;
    if (orow < ROWS_) {
      out[(size_t)orow * D_ + nn]      = fmaxf(c0[r] + bb0, 0.f);
      out[(size_t)orow * D_ + 16 + nn] = fmaxf(c1[r] + bb1, 0.f);
    }
  }
}

struct AP { const float *wq,*bq,*wk,*bk,*wv,*bv,*wo,*bo,*g,*be; };

extern "C" void kernel_launch(void* const* d_in, const int* in_sizes, int n_in,
                              void* d_out, int out_size, void* d_ws, size_t ws_size,
                              hipStream_t stream) {
  (void)in_sizes; (void)n_in; (void)out_size; (void)ws_size;
  int id = 0;
  auto F = [&]() -> const float* { return (const float*)d_in[id++]; };
  const float* x    = F();
  const float* adj1 = F();
  const float* adj2 = F();
  const float* emb_w1 = F(); const float* emb_b1 = F();
  const float* emb_w2 = F(); const float* emb_b2 = F();
  AP tp[2], sp[2];
  for (int i = 0; i < 2; ++i) { AP& p = tp[i];
    p.wq=F(); p.bq=F(); p.wk=F(); p.bk=F(); p.wv=F(); p.bv=F(); p.wo=F(); p.bo=F(); p.g=F(); p.be=F(); }
  for (int i = 0; i < 2; ++i) { AP& p = sp[i];
    p.wq=F(); p.bq=F(); p.wk=F(); p.bk=F(); p.wv=F(); p.bv=F(); p.wo=F(); p.bo=F(); p.g=F(); p.be=F(); }
  const float* fus_w1 = F(); const float* fus_b1 = F();
  const float* fus_w2 = F(); const float* fus_b2 = F();
  const float* gate_w1 = F(); const float* gate_b1 = F();
  const float* gate_w2 = F(); const float* gate_b2 = F();
  const float* tcn_w[3]; tcn_w[0]=F(); tcn_w[1]=F(); tcn_w[2]=F();
  const float* tcn_b[3]; tcn_b[0]=F(); tcn_b[1]=F(); tcn_b[2]=F();
  const float* tcn_fw = F(); const float* tcn_fb = F();
  const float* out_w = F();  const float* out_b = F();

  float* ws = (float*)d_ws;
  const size_t NB = (size_t)ROWS_ * D_;
  float* e    = ws + 0*NB;
  float* bufA = ws + 1*NB;
  float* bufB = ws + 2*NB;
  float* qb   = ws + 3*NB;
  float* kbuf = ws + 4*NB;
  float* vb   = ws + 5*NB;
  float* ob   = ws + 6*NB;
  float* hdt  = ws + 7*NB;
  float* hdg  = ws + 8*NB;
  float* hda  = ws + 9*NB;
  float* hds  = ws + 10*NB;

  const int lb  = (ROWS_ + 15) / 16;          // 975 row-tiles
  const int eb  = (ROWS_ * D_ + 255) / 256;   // elementwise blocks

  auto attn_layer = [&](const float* xin, float* xout, const AP& p,
                        const float* adj, int BB, int S, int causal) {
    k_lin<32,0,0><<<lb, 32, 0, stream>>>(xin, nullptr, p.wq, p.bq, qb,   ROWS_);
    k_lin<32,0,0><<<lb, 32, 0, stream>>>(xin, nullptr, p.wk, p.bk, kbuf, ROWS_);
    k_lin<32,0,0><<<lb, 32, 0, stream>>>(xin, nullptr, p.wv, p.bv, vb,   ROWS_);
    dim3 g((S + 15) / 16, H_, BB);
    if (causal) k_attn<0,1><<<g, 32, 0, stream>>>(qb, kbuf, vb, ob, adj, S);
    else        k_attn<1,0><<<g, 32, 0, stream>>>(qb, kbuf, vb, ob, adj, S);
    k_projln<<<lb, 32, 0, stream>>>(xin, ob, p.wo, p.bo, p.g, p.be, xout, ROWS_);
  };

  // 1) embedding
  k_embed<<<(ROWS_ + 127) / 128, 128, 0, stream>>>(x, emb_w1, emb_b1, emb_w2, emb_b2, e);

  // 2) temporal attention on (B*N, T, 32), causal
  k_btn_to_bnt<<<eb, 256, 0, stream>>>(e, bufA);
  attn_layer(bufA, bufB, tp[0], nullptr, B_ * N_, T_, 1);
  attn_layer(bufB, bufA, tp[1], nullptr, B_ * N_, T_, 1);
  k_bnt_to_btn<<<eb, 256, 0, stream>>>(bufA, hdt);

  // 3) spatial attention on (B*T, N, 32) with adj1 / adj2
  attn_layer(e,    bufA, sp[0], adj1, B_ * T_, N_, 0);
  attn_layer(bufA, hdg,  sp[1], adj1, B_ * T_, N_, 0);
  attn_layer(e,    bufB, sp[0], adj2, B_ * T_, N_, 0);
  attn_layer(bufB, hda,  sp[1], adj2, B_ * T_, N_, 0);

  // 4) fusion: hds = lin(relu(lin(concat(hdg,hda))))
  k_lin<64,1,0><<<lb, 32, 0, stream>>>(hdg, hda, fus_w1, fus_b1, bufA, ROWS_);
  k_lin<32,0,0><<<lb, 32, 0, stream>>>(bufA, nullptr, fus_w2, fus_b2, hds, ROWS_);

  // 5) gate: z = lin(sigmoid(lin(concat(hdt,hds)))); hst = z*hdt + (1-z)*hds
  k_lin<64,2,0><<<lb, 32, 0, stream>>>(hdt, hds, gate_w1, gate_b1, bufB, ROWS_);
  k_lin<32,0,0><<<lb, 32, 0, stream>>>(bufB, nullptr, gate_w2, gate_b2, bufA, ROWS_);
  k_gate<<<eb, 256, 0, stream>>>(bufA, hdt, hds, bufB);

  // 6) TCN: 3 causal dilated convs (dil 1,2,4), channel-last
  k_conv<<<lb, 32, 0, stream>>>(bufB, tcn_w[0], tcn_b[0], qb,   1);
  k_conv<<<lb, 32, 0, stream>>>(qb,   tcn_w[1], tcn_b[1], kbuf, 2);
  k_conv<<<lb, 32, 0, stream>>>(kbuf, tcn_w[2], tcn_b[2], vb,   4);

  // 7) final projections on the t=T-1 slice only: (B*N, 32) rows
  const int rows2 = B_ * N_;
  const int lb2 = (rows2 + 15) / 16;
  k_lin<32,0,1><<<lb2, 32, 0, stream>>>(vb, nullptr, tcn_fw, tcn_fb, ob, rows2);
  k_lin<32,0,0><<<lb2, 32, 0, stream>>>(ob, nullptr, out_w, out_b, (float*)d_out, rows2);
}